// LinearCRF_52518860095929
// MI455X (gfx1250) — compile-verified
//
#include <hip/hip_runtime.h>
#include <stdint.h>

// Viterbi / linear-CRF decode for MI455X (gfx1250, wave32).
// Max-plus recurrence is not WMMA-expressible (tropical semiring, argmax
// needed); CDNA5-specific paths used instead:
//  - 320KB LDS/WGP: full 264KB transposed transition matrix resident per WGP
//    (stride 258 floats -> conflict-free wave32 b64 loads, 8B aligned rows)
//  - async global->LDS emission-row prefetch (global_load_async_to_lds_b32 +
//    s_wait_asynccnt) double-buffered across timesteps
//  - mask row staged in LDS: zero global accesses in the 1023-step hot loop
//    except the coalesced 1B-per-thread backpointer store.

#define Bb   64
#define Tt   1024
#define Ll   256
#define PADl 0
#define BOSl 1
#define EOSl 2
#define TSTRIDE 258   // padded row stride (floats): bank (2*lane+i)%64 -> conflict-free

// Generic pointer (known to point into LDS) -> 32-bit LDS byte offset.
__device__ __forceinline__ unsigned lds_addr_of(const void* p) {
  return (unsigned)(unsigned long long)
      (__attribute__((address_space(3))) const char*)p;
}

__device__ __forceinline__ void async_load_b32_to_lds(unsigned lds_off, const void* gsrc) {
  unsigned long long ga = (unsigned long long)(uintptr_t)gsrc;
  asm volatile("global_load_async_to_lds_b32 %0, %1, off"
               :: "v"(lds_off), "v"(ga) : "memory");
}

__device__ __forceinline__ void wait_async_zero() {
  asm volatile("s_wait_asynccnt 0x0" ::: "memory");
}

__global__ __launch_bounds__(Ll, 1)
void viterbi_forward(const float* __restrict__ x,
                     const float* __restrict__ trans,
                     const uint8_t* __restrict__ mask,
                     uint8_t* __restrict__ bp,
                     int* __restrict__ best_last,
                     float* __restrict__ best_score) {
  extern __shared__ float smem[];
  float*    transT = smem;                      // Ll * TSTRIDE floats (264 KB)
  float*    score  = smem + Ll * TSTRIDE;       // 2 * Ll (double buffer)
  float*    xbuf   = score + 2 * Ll;            // 2 * Ll (double buffer)
  uint32_t* mbuf   = (uint32_t*)(xbuf + 2 * Ll); // Tt/4 u32 = 1KB mask row

  const int tid = threadIdx.x;                  // column j
  const int b   = blockIdx.x;                   // batch element
  const float* xrow0 = x + (size_t)b * Tt * Ll;

  // Kick off async prefetch of emission row t=1 first: it streams in behind
  // the 264KB transition staging below.
  async_load_b32_to_lds(lds_addr_of(&xbuf[Ll + tid]), xrow0 + Ll + tid);

  // Stage mask row (1KB) into LDS.
  mbuf[tid] = ((const uint32_t*)(mask + (size_t)b * Tt))[tid];

  // t = 0: score0 = trans[BOS,:] + x[b,0,:]
  score[tid] = trans[BOSl * Ll + tid] + xrow0[tid];

  // transT[j*TSTRIDE + i] = trans[i*Ll + j]  (coalesced global, conflict-free LDS)
  #pragma unroll 4
  for (int i = 0; i < Ll; ++i)
    transT[tid * TSTRIDE + i] = trans[i * Ll + tid];

  __syncthreads();

  const float*   trow = transT + tid * TSTRIDE;
  const uint8_t* mrow = (const uint8_t*)mbuf;
  int cur = 0;
  for (int t = 1; t < Tt; ++t) {
    wait_async_zero();                          // row t resident in xbuf[t&1]
    // Branch-free prefetch of next row (clamped; last clamp lands in the
    // unused buffer half).
    {
      const int tn = (t + 1 < Tt) ? (t + 1) : (Tt - 1);
      async_load_b32_to_lds(lds_addr_of(&xbuf[((t + 1) & 1) * Ll + tid]),
                            xrow0 + (size_t)tn * Ll + tid);
    }

    const float* sc = score + cur * Ll;
    float best = -INFINITY;
    int   bi   = 0;
    #pragma unroll 4
    for (int i = 0; i < Ll; i += 4) {
      const float4 s4 = *(const float4*)(sc + i);     // uniform addr -> LDS broadcast
      const float2 t0 = *(const float2*)(trow + i);   // per-thread row, conflict-free
      const float2 t1 = *(const float2*)(trow + i + 2);
      float c0 = s4.x + t0.x; if (c0 > best) { best = c0; bi = i;     }
      float c1 = s4.y + t0.y; if (c1 > best) { best = c1; bi = i + 1; }
      float c2 = s4.z + t1.x; if (c2 > best) { best = c2; bi = i + 2; }
      float c3 = s4.w + t1.y; if (c3 > best) { best = c3; bi = i + 3; }
    }

    const float emit = xbuf[(t & 1) * Ll + tid];
    const bool  m    = mrow[t] != 0;                  // LDS broadcast, no global
    const float ns   = m ? (best + emit) : sc[tid];
    const int   nb   = m ? bi : tid;

    bp[((size_t)(b * Tt + t)) * Ll + tid] = (uint8_t)nb;
    score[(cur ^ 1) * Ll + tid] = ns;
    __syncthreads();
    cur ^= 1;
  }

  // final = score + trans[:,EOS];  trans[j][EOS] = transT[EOS*TSTRIDE + j]
  const float fin = score[cur * Ll + tid] + transT[EOSl * TSTRIDE + tid];
  score[(cur ^ 1) * Ll + tid] = fin;
  __syncthreads();

  if (tid == 0) {                                // ascending strict-> scan = jnp.argmax ties
    const float* f = score + (cur ^ 1) * Ll;
    float bv = f[0]; int bi2 = 0;
    for (int j = 1; j < Ll; ++j)
      if (f[j] > bv) { bv = f[j]; bi2 = j; }
    best_score[b] = bv;
    best_last[b]  = bi2;
  }
}

__global__ void viterbi_backtrack(const uint8_t* __restrict__ bp,
                                  const uint8_t* __restrict__ mask,
                                  const int* __restrict__ best_last,
                                  int* __restrict__ path) {
  const int b = blockIdx.x * blockDim.x + threadIdx.x;
  if (b >= Bb) return;
  int idx = best_last[b];
  for (int t = Tt - 1; t >= 1; --t) {
    path[b * Tt + t] = mask[b * Tt + t] ? idx : PADl;
    idx = bp[((size_t)(b * Tt + t)) * Ll + idx];  // 16MB bp stays hot in 192MB L2
  }
  path[b * Tt] = mask[b * Tt] ? idx : PADl;
}

extern "C" void kernel_launch(void* const* d_in, const int* in_sizes, int n_in,
                              void* d_out, int out_size, void* d_ws, size_t ws_size,
                              hipStream_t stream) {
  const float*   x     = (const float*)d_in[0];     // (B,T,L) f32
  const float*   trans = (const float*)d_in[1];     // (L,L)   f32
  const uint8_t* mask  = (const uint8_t*)d_in[2];   // (B,T)   bool (1 byte)

  int*   path       = (int*)d_out;                      // (B,T) int32 bits
  float* best_score = (float*)d_out + (size_t)Bb * Tt;  // (B,)  f32

  uint8_t* bp        = (uint8_t*)d_ws;                              // B*T*L bytes
  int*     best_last = (int*)((char*)d_ws + (size_t)Bb * Tt * Ll);  // B ints

  const size_t smem_bytes =
      (size_t)(Ll * TSTRIDE + 4 * Ll + Tt / 4) * sizeof(float);     // ~269 KB

  viterbi_forward<<<Bb, Ll, smem_bytes, stream>>>(x, trans, mask, bp,
                                                  best_last, best_score);
  viterbi_backtrack<<<2, 32, 0, stream>>>(bp, mask, best_last, path);
}